// Noiser_72103910966019
// MI455X (gfx1250) — compile-verified
//
#include <hip/hip_runtime.h>
#include <hip/hip_bf16.h>

// ---------------- problem constants (match reference) ----------------
constexpr int kB         = 32768;          // samples
constexpr int kTotal     = kB * 128;       // 4,194,304 atoms
constexpr int kTotalF    = kTotal * 3;     // 12,582,912 floats per (SumN,3) array
constexpr int kLElems    = kB * 9;         // l_mid floats
constexpr int kEElems    = kB;             // e_mid floats
// DELTA = 1/(TIMESTEPS-1) as float32, divide (not mul-by-recip) to mirror XLA
#define DELTA_F ((float)(1.0 / 999.0))

#define AS1 __attribute__((address_space(1)))
#define AS3 __attribute__((address_space(3)))

typedef int   v4i __attribute__((ext_vector_type(4)));
typedef float v4f __attribute__((ext_vector_type(4)));

#if __has_builtin(__builtin_amdgcn_global_load_async_to_lds_b128) && \
    __has_builtin(__builtin_amdgcn_global_store_async_from_lds_b128)
#define HAVE_ASYNC_LDS 1
#else
#define HAVE_ASYNC_LDS 0
#endif

static __device__ __forceinline__ void wait_async0() {
#if __has_builtin(__builtin_amdgcn_s_wait_asynccnt)
  __builtin_amdgcn_s_wait_asynccnt(0);
#else
  asm volatile("s_wait_asynccnt 0" ::: "memory");
#endif
}

// ---------------- kernel 1: exclusive scan of num_atoms -> atom offsets ----
// B = 32768 = 1024 threads x 32 elements each. One block.
__global__ __launch_bounds__(1024) void scan_kernel(const int* __restrict__ na,
                                                    int* __restrict__ offs) {
  __shared__ int ps[1024];
  const int tid  = threadIdx.x;
  const int base = tid * 32;
  int s = 0;
#pragma unroll
  for (int i = 0; i < 32; ++i) s += na[base + i];
  ps[tid] = s;
  __syncthreads();
  // Hillis-Steele inclusive scan over the 1024 partials
  for (int d = 1; d < 1024; d <<= 1) {
    int v = (tid >= d) ? ps[tid - d] : 0;
    __syncthreads();
    ps[tid] += v;
    __syncthreads();
  }
  int run = ps[tid] - s;  // exclusive prefix at the start of this chunk
#pragma unroll
  for (int i = 0; i < 32; ++i) {
    offs[base + i] = run;
    run += na[base + i];
  }
}

// ---------------- kernel 2: fused ragged-gather noiser ----------------
// One workgroup per sample b. All atoms of a sample share one noise scale,
// so the block's float range [3*off, 3*(off+n)) is contiguous. The range is
// only 4B-aligned, so: scalar head (<=3), 16B-aligned float4 body (b128
// loads/stores), scalar tail (<=3). Outputs stored non-temporal so the 144MB
// of read streams can stay resident in the 192MB L2 across graph replays.
__global__ __launch_bounds__(256) void noiser_main(
    const float* __restrict__ x_mid,
    const float* __restrict__ x_prev,
    const float* __restrict__ noise_x,
    const float* __restrict__ noise_scales,
    const int*   __restrict__ t,
    const int*   __restrict__ num_atoms,
    const int*   __restrict__ offs,
    float* __restrict__ out_xt,
    float* __restrict__ out_xtar) {
  const int    b    = blockIdx.x;
  const int    tid  = threadIdx.x;
  const int    n3   = num_atoms[b] * 3;        // 96 .. 672
  const size_t fbeg = (size_t)offs[b] * 3;
  const size_t fend = fbeg + (size_t)n3;
  const float  s    = noise_scales[t[b]];

  // hint the three read streams into cache early (gfx1250 global_prefetch_b8)
  __builtin_prefetch(&x_mid[fbeg],   0, 3);
  __builtin_prefetch(&noise_x[fbeg], 0, 3);
  __builtin_prefetch(&x_prev[fbeg],  0, 3);

  const size_t abeg = (fbeg + 3) & ~(size_t)3;  // first 16B-aligned float
  const size_t aend = fend & ~(size_t)3;        // end of aligned region

  // scalar head + tail (each at most 3 elements)
  {
    const size_t kh = fbeg + (size_t)tid;
    if (kh < abeg) {
      const float xt = fmaf(s, noise_x[kh], x_mid[kh]);
      __builtin_nontemporal_store(xt, &out_xt[kh]);
      __builtin_nontemporal_store((x_prev[kh] - xt) / DELTA_F, &out_xtar[kh]);
    }
    const size_t kt = aend + (size_t)tid;
    if (kt < fend) {
      const float xt = fmaf(s, noise_x[kt], x_mid[kt]);
      __builtin_nontemporal_store(xt, &out_xt[kt]);
      __builtin_nontemporal_store((x_prev[kt] - xt) / DELTA_F, &out_xtar[kt]);
    }
  }

  // vector body: float4 indices [abeg/4, aend/4), <=168 per block -> 1 pass
  const v4f* __restrict__ xm4 = (const v4f*)x_mid;
  const v4f* __restrict__ xp4 = (const v4f*)x_prev;
  const v4f* __restrict__ nx4 = (const v4f*)noise_x;
  v4f* __restrict__ ot4 = (v4f*)out_xt;
  v4f* __restrict__ oa4 = (v4f*)out_xtar;
  for (size_t v = (abeg >> 2) + (size_t)tid; v < (aend >> 2); v += 256) {
    const v4f xm = xm4[v];
    const v4f nx = nx4[v];
    const v4f xp = xp4[v];
    const v4f xt  = xm + s * nx;               // contracts to v_fma
    const v4f xta = (xp - xt) / DELTA_F;       // IEEE divide, mirrors XLA
    __builtin_nontemporal_store(xt,  &ot4[v]);
    __builtin_nontemporal_store(xta, &oa4[v]);
  }
}

// ---------------- kernel 3: pass-through copies via async LDS DMA ----------
// Each block copies 1024 floats (4KB): per-lane b128 global->LDS async load,
// s_wait_asynccnt, then b128 LDS->global async store. Waves only touch their
// own LDS slice, so no barrier is needed. Falls back to a plain b128 copy if
// this toolchain doesn't declare the async builtins.
__global__ __launch_bounds__(256) void copy4k_kernel(const float* __restrict__ src,
                                                     float* __restrict__ dst) {
  const size_t base = ((size_t)blockIdx.x * 256 + threadIdx.x) * 4;
#if HAVE_ASYNC_LDS
  __shared__ v4i buf[256];
  AS1 v4i* gsrc = (AS1 v4i*)(src + base);
  AS1 v4i* gdst = (AS1 v4i*)(dst + base);
  AS3 v4i* lptr = (AS3 v4i*)(&buf[threadIdx.x]);
  __builtin_amdgcn_global_load_async_to_lds_b128(gsrc, lptr, 0, 0);
  wait_async0();  // LDS write visible before the DMA store reads it
  __builtin_amdgcn_global_store_async_from_lds_b128(gdst, lptr, 0, 0);
  wait_async0();
#else
  const float4 v = *(const float4*)(src + base);
  *(float4*)(dst + base) = v;
#endif
}

// ---------------- launcher ----------------
extern "C" void kernel_launch(void* const* d_in, const int* in_sizes, int n_in,
                              void* d_out, int out_size, void* d_ws, size_t ws_size,
                              hipStream_t stream) {
  (void)in_sizes; (void)n_in; (void)out_size; (void)ws_size;

  const float* l_mid        = (const float*)d_in[0];   // (B,3,3)
  const float* x_mid        = (const float*)d_in[1];   // (SumN,3)
  const float* x_mid_prev   = (const float*)d_in[2];   // (SumN,3)
  const float* e_mid        = (const float*)d_in[3];   // (B,)
  const float* noise_x      = (const float*)d_in[4];   // (SumN,3)
  const float* noise_scales = (const float*)d_in[5];   // (1000,)
  const int*   t            = (const int*)d_in[6];     // (B,)
  const int*   num_atoms    = (const int*)d_in[7];     // (B,)

  float* out      = (float*)d_out;
  float* out_xt   = out;                       // (SumN,3)
  float* out_xtar = out + kTotalF;             // (SumN,3)
  float* out_l    = out + 2 * kTotalF;         // (B,3,3)
  float* out_e    = out_l + kLElems;           // (B,)

  int* offs = (int*)d_ws;                      // kB ints of scratch

  scan_kernel<<<1, 1024, 0, stream>>>(num_atoms, offs);
  noiser_main<<<kB, 256, 0, stream>>>(x_mid, x_mid_prev, noise_x, noise_scales,
                                      t, num_atoms, offs, out_xt, out_xtar);
  copy4k_kernel<<<kLElems / 1024, 256, 0, stream>>>(l_mid, out_l);  // 288 blocks
  copy4k_kernel<<<kEElems / 1024, 256, 0, stream>>>(e_mid, out_e);  // 32 blocks
}